// LocBlock_12463995093295
// MI455X (gfx1250) — compile-verified
//
#include <hip/hip_runtime.h>
#include <math.h>

// CDNA5 wave32 WMMA vector types
typedef float v2f __attribute__((ext_vector_type(2)));
typedef float v8f __attribute__((ext_vector_type(8)));
// async-to-LDS builtin operand type: 16-byte int vector
typedef int vi4 __attribute__((vector_size(16)));
typedef __attribute__((address_space(1))) vi4* gptr128;
typedef __attribute__((address_space(3))) vi4* lptr128;

#define WAVES   8
#define TPB     256
#define TILE    128    // points staged per wave per tile (32 lanes x float4)
#define FSTRIDE 132    // padded row stride: 16B-aligned rows, conflict-free banks
#define FROWS   8      // rows 0..5 = m,y; 6 = ones; 7 = zeros (rows 8..15 alias row 7)

#if __has_builtin(__builtin_amdgcn_global_load_async_to_lds_b128)
#define USE_ASYNC_LDS 1
#else
#define USE_ASYNC_LDS 0
#endif

__global__ __launch_bounds__(TPB) void wgram_qassemble_kernel(
    const float* __restrict__ m,   // (B,3,N)
    const float* __restrict__ y,   // (B,3,N)
    const float* __restrict__ w,   // (B,1,N)
    float* __restrict__ out,       // [B*169 Qn | B scales | B offsets]
    int B, int N)
{
    __shared__ __align__(16) float feat[WAVES][FROWS * FSTRIDE];
    __shared__ __align__(16) float wtile[WAVES][TILE];
    __shared__ float redbuf[WAVES * 256];
    __shared__ float qbuf[176]; // 169 Q entries + scale slot

    const int b    = blockIdx.x;
    const int t    = threadIdx.x;
    const int wv   = t >> 5;
    const int lane = t & 31;
    const int r    = lane & 15;          // logical feature row (A.M == B.N for this lane)
    const int rr   = (r < 8) ? r : 7;    // rows 8..15 are zero -> broadcast the zero row
    const int kb   = (lane >> 4) << 1;   // K base: lanes 0-15 -> K{0,1}, lanes 16-31 -> K{2,3}

    // constant rows (per-wave region; ordered by the barrier below)
    for (int c = lane; c < FSTRIDE; c += 32) {
        feat[wv][6 * FSTRIDE + c] = 1.0f;   // ones row
        feat[wv][7 * FSTRIDE + c] = 0.0f;   // zero row (shared by logical rows 7..15)
    }

    const float* mB = m + (size_t)b * 3 * N;
    const float* yB = y + (size_t)b * 3 * N;
    const float* wB = w + (size_t)b * N;

    v8f acc = {0.f, 0.f, 0.f, 0.f, 0.f, 0.f, 0.f, 0.f};

    const int ntiles = N / (WAVES * TILE);   // assumes N % 1024 == 0 (here N = 4096)
    for (int tl = 0; tl < ntiles; ++tl) {
        const int p0 = (tl * WAVES + wv) * TILE;
        __syncthreads();
#if USE_ASYNC_LDS
        // direct global -> LDS, one b128 per lane per row (512B per instruction)
        #pragma unroll
        for (int r6 = 0; r6 < 6; ++r6) {
            const float* src = (r6 < 3 ? mB + r6 * N : yB + (r6 - 3) * N) + p0 + 4 * lane;
            __builtin_amdgcn_global_load_async_to_lds_b128(
                (gptr128)(src),
                (lptr128)(&feat[wv][r6 * FSTRIDE + 4 * lane]),
                0, 0);
        }
        __builtin_amdgcn_global_load_async_to_lds_b128(
            (gptr128)(wB + p0 + 4 * lane),
            (lptr128)(&wtile[wv][4 * lane]),
            0, 0);
#if __has_builtin(__builtin_amdgcn_s_wait_asynccnt)
        __builtin_amdgcn_s_wait_asynccnt(0);
#else
        asm volatile("s_wait_asynccnt 0x0" ::: "memory");
#endif
#else
        // fallback: float4 through VGPRs (compiler merges LDS stores to b128)
        #pragma unroll
        for (int r6 = 0; r6 < 6; ++r6) {
            const float* src = (r6 < 3 ? mB + r6 * N : yB + (r6 - 3) * N) + p0;
            float4 v = reinterpret_cast<const float4*>(src)[lane];
            feat[wv][r6 * FSTRIDE + 4 * lane]     = v.x;
            feat[wv][r6 * FSTRIDE + 4 * lane + 1] = v.y;
            feat[wv][r6 * FSTRIDE + 4 * lane + 2] = v.z;
            feat[wv][r6 * FSTRIDE + 4 * lane + 3] = v.w;
        }
        {
            float4 v = reinterpret_cast<const float4*>(wB + p0)[lane];
            wtile[wv][4 * lane]     = v.x;
            wtile[wv][4 * lane + 1] = v.y;
            wtile[wv][4 * lane + 2] = v.z;
            wtile[wv][4 * lane + 3] = v.w;
        }
#endif
        __syncthreads();
        // 32 WMMA steps of K=4 points: G += (w .* X)(16x4) * X^T(4x16)
        #pragma unroll
        for (int s = 0; s < TILE / 4; ++s) {
            const int p = s * 4 + kb;
            float bx = feat[wv][rr * FSTRIDE + p];
            float by = feat[wv][rr * FSTRIDE + p + 1];
            float wx = wtile[wv][p];
            float wy = wtile[wv][p + 1];
            v2f A  = { wx * bx, wy * by };
            v2f Bv = { bx, by };
            acc = __builtin_amdgcn_wmma_f32_16x16x4_f32(
                      false, A, false, Bv, (short)0, acc, false, false);
        }
    }
    __syncthreads();

    // cross-wave reduction of the 16x16 accumulators through LDS
    #pragma unroll
    for (int v = 0; v < 8; ++v)
        redbuf[wv * 256 + v * 32 + lane] = acc[v];
    __syncthreads();
    float g = 0.f;
    #pragma unroll
    for (int wvi = 0; wvi < WAVES; ++wvi)
        g += redbuf[wvi * 256 + t];
    // column t is touched only by thread t -> safe read-then-write
    redbuf[t] = g;                 // G[M][Nn] at redbuf[M*32 + Nn] (M,Nn <= 6)
    if (t < 176) qbuf[t] = 0.f;    // Q is mostly zeros
    __syncthreads();

    if (t == 0) {
        float Mw[3][3], MY[3][3], mw[3], yw[3];
        #pragma unroll
        for (int i = 0; i < 3; ++i) {
            #pragma unroll
            for (int j = 0; j < 3; ++j) {
                Mw[i][j] = redbuf[i * 32 + j];
                MY[i][j] = redbuf[i * 32 + 3 + j];
            }
            mw[i] = redbuf[i * 32 + 6];
            yw[i] = redbuf[(3 + i) * 32 + 6];
        }
        const float sw = redbuf[6 * 32 + 6];
        const float yy = redbuf[3 * 32 + 3] + redbuf[4 * 32 + 4] + redbuf[5 * 32 + 5];

        float ss = 3.f * sw * sw;
        #pragma unroll
        for (int i = 0; i < 3; ++i) {
            ss += 6.f * mw[i] * mw[i] + 2.f * yw[i] * yw[i];
            #pragma unroll
            for (int j = 0; j < 3; ++j) {
                ss += 3.f * Mw[i][j] * Mw[i][j] + 2.f * MY[i][j] * MY[i][j];
                #pragma unroll
                for (int k = 0; k < 3; ++k)   // Qcc: Q[1+3i+k][1+3j+k] = Mw[i][j]
                    qbuf[(1 + 3 * i + k) * 13 + (1 + 3 * j + k)] = Mw[i][j];
                const float v = -MY[i][j];    // Qch: col 0 and row 0
                qbuf[(1 + 3 * i + j) * 13] = v;
                qbuf[1 + 3 * i + j]        = v;
            }
            #pragma unroll
            for (int k = 0; k < 3; ++k) {     // Qct and its transpose
                const float v = -mw[i];
                qbuf[(1 + 3 * i + k) * 13 + (10 + k)] = v;
                qbuf[(10 + k) * 13 + (1 + 3 * i + k)] = v;
            }
        }
        #pragma unroll
        for (int k = 0; k < 3; ++k) {         // Qtt diag + yw borders
            qbuf[(10 + k) * 13 + (10 + k)] = sw;
            qbuf[(10 + k) * 13]            = yw[k];
            qbuf[10 + k]                   = yw[k];
        }
        const float scl = sqrtf(ss);
        qbuf[169] = scl;
        out[(size_t)B * 169 + b] = scl;   // scales
        out[(size_t)B * 170 + b] = yy;    // offsets
    }
    __syncthreads();

    const float scl = qbuf[169];
    if (t < 169)
        out[(size_t)b * 169 + t] = qbuf[t] / scl;
}

extern "C" void kernel_launch(void* const* d_in, const int* in_sizes, int n_in,
                              void* d_out, int out_size, void* d_ws, size_t ws_size,
                              hipStream_t stream) {
    const float* m = (const float*)d_in[0];
    const float* y = (const float*)d_in[1];
    const float* w = (const float*)d_in[2];
    float* out = (float*)d_out;

    const int B = out_size / 171;        // 169 (Qn) + 1 (scale) + 1 (offset) per batch
    const int N = in_sizes[2] / B;       // weights are (B,1,N)

    wgram_qassemble_kernel<<<dim3(B), dim3(TPB), 0, stream>>>(m, y, w, out, B, N);
}